// TrainableButterfly_360777252984
// MI455X (gfx1250) — compile-verified
//
#include <hip/hip_runtime.h>
#include <hip/hip_bf16.h>

// CDNA5 (gfx1250) trainable-butterfly kernel.
// Design: one 1024-thread workgroup owns a 16-row batch tile, keeps all
// 2048 complex values (256 KB) resident in LDS across all 11 butterfly
// levels, and maps each level's 1024 independent 2x2-complex (= 4x4 real)
// butterflies onto V_WMMA_F32_16X16X4_F32: a 16x16 block-diagonal W
// (4 pairs) times a 16-component x 16-batch tile, as 4 chained K=4 WMMAs.
// Inter-level crossing permutations (identity complex factor since
// CROSS_T=1, CROSS_PHI=0) are fused into the next level's LDS reads.
// Global traffic is the roofline minimum: 64 MiB in + 64 MiB out, both
// streamed non-temporally (no reuse).

typedef __attribute__((ext_vector_type(2))) float v2f;
typedef __attribute__((ext_vector_type(8))) float v8f;

#define LENGTH   2048
#define NLEVEL   11
#define BATCH    4096
#define BTILE    16                 // batch columns per workgroup (WMMA N)
#define NTHREADS 1024               // 32 waves (wave32)
#define NWAVES   (NTHREADS / 32)
#define NGROUPS  (2 * LENGTH / 16)  // 256 groups of 16 components
#define GPW      (NGROUPS / NWAVES) // 8 groups per wave

// Input permutation feeding butterfly level `lev`:
//   lev==0 : identity (bit-reversal already applied at global->LDS load)
//   lev>=1 : FWD_IDX[lev-1], block size bs = 2^(lev+1):
//            src(blk*bs + 2i + j) = blk*bs + j*bs/2 + i
__device__ __forceinline__ int perm_e(int lev, int e) {
  if (lev == 0) return e;
  const int bs  = 1 << (lev + 1);
  const int blk = e & ~(bs - 1);
  const int r   = e & (bs - 1);
  return blk | ((r & 1) << lev) | (r >> 1);
}

__global__ __launch_bounds__(NTHREADS)
void butterfly_wmma_kernel(const float* __restrict__ x,
                           const float* __restrict__ phases,
                           float* __restrict__ out) {
  // comp-major data tile: sdata[comp (2*elem+reim)][batch col] : 256 KB
  __shared__ float sdata[2 * LENGTH * BTILE];
  // per-level 4x4 real butterfly matrices, row-major: 1024 x 16 : 64 KB
  // (reused at the end as the 2048 x (cos,sin) output-phase table)
  __shared__ float sw[(LENGTH / 2) * 16];

  const int tid   = threadIdx.x;
  const int wv    = tid >> 5;
  const int lane  = tid & 31;
  const int col   = lane & 15;  // WMMA N column (batch) / A row m
  const int hi    = lane >> 4;  // 0 -> holds K 0/1, 1 -> holds K 2/3
  const int mrow  = col & 3;    // row inside 4x4 block
  const int mpair = col >> 2;   // which local pair this lane's A-row is in
  const int b0    = blockIdx.x * BTILE;

  // ---- global -> LDS, fusing bit reversal (x[:, BITREV, :]) ----
  // read contiguous source element e (coalesced, non-temporal float2),
  // scatter to component rows of brev(e) (BITREV is an involution).
  for (int t = tid; t < BTILE * LENGTH; t += NTHREADS) {
    const int b = t >> 11;              // 0..15
    const int e = t & (LENGTH - 1);     // 0..2047
    const v2f v = __builtin_nontemporal_load(
        (const v2f*)(x + (((size_t)(b0 + b)) * LENGTH + e) * 2));
    const int be = (int)(__brev((unsigned)e) >> 21);  // 11-bit reversal
    sdata[(2 * be    ) * BTILE + b] = v.x;
    sdata[(2 * be + 1) * BTILE + b] = v.y;
  }

  for (int lev = 0; lev < NLEVEL; ++lev) {
    // ---- build this level's 4x4 real matrices in LDS ----
    // wr = [[c1,-s2],[-s1,c2]], wi = [[s1,c2],[c1,s2]]; acting on
    // interleaved (re0,im0,re1,im1):
    {
      const int p = tid;  // 1024 pairs, 1024 threads
      const float2 ph =
          *(const float2*)(phases + (size_t)lev * LENGTH + 2 * p);
      float s1, c1, s2, c2;
      __sincosf(ph.x, &s1, &c1);
      __sincosf(ph.y, &s2, &c2);
      float4* w4 = (float4*)&sw[p * 16];
      w4[0] = make_float4( c1, -s1, -s2, -c2);   // y0.re
      w4[1] = make_float4( s1,  c1,  c2, -s2);   // y0.im
      w4[2] = make_float4(-s1, -c1,  c2, -s2);   // y1.re
      w4[3] = make_float4( c1, -s1,  s2,  c2);   // y1.im
    }
    __syncthreads();

    // ---- phase 1: all reads + WMMA compute into registers ----
    v8f acc[GPW];
#pragma unroll
    for (int gi = 0; gi < GPW; ++gi) {
      const int g = wv * GPW + gi;
      // This lane's two A-matrix entries (row m=col, cols 2*hi..2*hi+1 of
      // the 4x4 block it belongs to). Nonzero only in K-slice k == mpair.
      const v2f w2 =
          *(const v2f*)&sw[(g * 4 + mpair) * 16 + mrow * 4 + 2 * hi];
      v8f c = {};
#pragma unroll
      for (int k = 0; k < 4; ++k) {
        v2f a;
        a.x = (mpair == k) ? w2.x : 0.0f;
        a.y = (mpair == k) ? w2.y : 0.0f;
        // B slice k rows = components 16g+4k .. 16g+4k+3; this lane holds
        // rows 4k+2*hi (v0) and 4k+2*hi+1 (v1), column `col`. Both rows
        // share element index e; apply the fused crossing permutation.
        const int e  = 8 * g + 2 * k + hi;
        const int se = perm_e(lev, e);
        v2f b;
        b.x = sdata[(2 * se    ) * BTILE + col];
        b.y = sdata[(2 * se + 1) * BTILE + col];
        // D = A(16x4) * B(4x16) + C   (fp32 WMMA, wave32)
        c = __builtin_amdgcn_wmma_f32_16x16x4_f32(
            false, a, false, b, (short)0, c, false, false);
      }
      acc[gi] = c;
    }
    __syncthreads();  // all reads of this level done before any write

    // ---- phase 2: write D tiles back in natural component order ----
    // D layout: VGPR j -> row j (lanes 0-15) / row j+8 (lanes 16-31)
#pragma unroll
    for (int gi = 0; gi < GPW; ++gi) {
      const int g = wv * GPW + gi;
#pragma unroll
      for (int j = 0; j < 8; ++j) {
        sdata[(16 * g + j + 8 * hi) * BTILE + col] = acc[gi][j];
      }
    }
    __syncthreads();
  }

  // ---- build the 2048-entry output phase table (cos,sin) in sw ----
  // (sw is free after the last level; avoids 16 redundant sincos per elem)
  for (int e = tid; e < LENGTH; e += NTHREADS) {
    float s, c;
    __sincosf(phases[(size_t)NLEVEL * LENGTH + e], &s, &c);
    sw[2 * e    ] = c;
    sw[2 * e + 1] = s;
  }
  __syncthreads();

  // ---- final elementwise phase multiply + coalesced non-temporal store ----
  for (int t = tid; t < BTILE * LENGTH; t += NTHREADS) {
    const int b = t >> 11;
    const int e = t & (LENGTH - 1);
    const float re = sdata[(2 * e    ) * BTILE + b];
    const float im = sdata[(2 * e + 1) * BTILE + b];
    const float c  = sw[2 * e    ];
    const float s  = sw[2 * e + 1];
    v2f o;
    o.x = re * c - im * s;
    o.y = re * s + im * c;
    __builtin_nontemporal_store(
        o, (v2f*)(out + (((size_t)(b0 + b)) * LENGTH + e) * 2));
  }
}

extern "C" void kernel_launch(void* const* d_in, const int* in_sizes, int n_in,
                              void* d_out, int out_size, void* d_ws,
                              size_t ws_size, hipStream_t stream) {
  (void)in_sizes; (void)n_in; (void)out_size; (void)d_ws; (void)ws_size;
  const float* x      = (const float*)d_in[0];
  const float* phases = (const float*)d_in[1];
  float* out          = (float*)d_out;
  butterfly_wmma_kernel<<<BATCH / BTILE, NTHREADS, 0, stream>>>(x, phases, out);
}